// AttentiveStylizationBlock_781684048586
// MI455X (gfx1250) — compile-verified
//
#include <hip/hip_runtime.h>
#include <math.h>

typedef float v2f __attribute__((ext_vector_type(2)));
typedef float v8f __attribute__((ext_vector_type(8)));

// ---- constants for this problem size ----
#define B_  8
#define T_  4096
#define E_  1024
#define M_  256
#define L_  512
#define EPS_ 1e-6f

__device__ __forceinline__ v8f vzero8() {
    v8f z;
    #pragma unroll
    for (int i = 0; i < 8; ++i) z[i] = 0.0f;
    return z;
}

// D = A(16x4) * B(4x16) + C, fp32 WMMA (CDNA5)
__device__ __forceinline__ v8f wmma4(v2f a, v2f b, v8f c) {
    return __builtin_amdgcn_wmma_f32_16x16x4_f32(
        /*neg_a=*/false, a, /*neg_b=*/false, b,
        /*c_mod=*/(short)0, c, /*reuse_a=*/false, /*reuse_b=*/false);
}

// ---------------------------------------------------------------------------
// Weight transpose: out[c][r] = in[r][c]   (so WMMA B-operand loads are b64)
// ---------------------------------------------------------------------------
__global__ __launch_bounds__(256) void k_transpose(
        const float* __restrict__ in, float* __restrict__ out, int R, int C) {
    __shared__ float t[32][33];
    int x = blockIdx.x * 32 + threadIdx.x;   // column in 'in'
    int y0 = blockIdx.y * 32;
    #pragma unroll
    for (int j = threadIdx.y; j < 32; j += 8) {
        int y = y0 + j;
        if (x < C && y < R) t[j][threadIdx.x] = in[(size_t)y * C + x];
    }
    __syncthreads();
    int ox = blockIdx.y * 32 + threadIdx.x;  // row in 'in' = col in out
    #pragma unroll
    for (int j = threadIdx.y; j < 32; j += 8) {
        int oy = blockIdx.x * 32 + j;        // col in 'in' = row in out
        if (ox < R && oy < C) out[(size_t)oy * R + ox] = t[threadIdx.x][j];
    }
}

// ---------------------------------------------------------------------------
// k = latent @ Wk + bk -> kout [B,M,E]  (row-major, K-dim contiguous for QK^T)
// v = latent @ Wv + bv -> vT  [B,E,M]   (stored TRANSPOSED so the pred GEMM's
//                                        B-operand is a contiguous b64 load)
// grid = B * (M/16) = 128 blocks, 256 threads (8 waves), wave owns 128 N-cols
// ---------------------------------------------------------------------------
#define KV_STRIDE 516   // 512 + 4 pad: conflict-free A-operand column reads
__global__ __launch_bounds__(256) void k_kv(
        const float* __restrict__ latent,
        const float* __restrict__ WkT, const float* __restrict__ bk,
        const float* __restrict__ WvT, const float* __restrict__ bv,
        float* __restrict__ kout, float* __restrict__ vT) {
    __shared__ float tile[16 * KV_STRIDE];   // 33KB
    const int b   = blockIdx.x >> 4;
    const int m0  = (blockIdx.x & 15) * 16;
    const int tid = threadIdx.x;
    const int lane = tid & 31;
    const int wv   = tid >> 5;
    const int l16  = lane & 15;
    const int hf   = lane >> 4;

    // stage latent[m0..m0+16, 0..512] into LDS
    const float* lb = latent + ((size_t)(b * M_ + m0)) * L_;
    #pragma unroll
    for (int it = 0; it < 32; ++it) {
        int flat = tid + it * 256;
        int r = flat >> 9, c = flat & 511;
        tile[r * KV_STRIDE + c] = lb[(size_t)r * L_ + c];
    }
    __syncthreads();

    // ---- k projection ----
    v8f acc[8];
    #pragma unroll
    for (int i = 0; i < 8; ++i) acc[i] = vzero8();
    {
        const float* bp0 = WkT + (size_t)(wv * 128 + l16) * L_;
        for (int kk = 0; kk < 128; ++kk) {
            v2f a = *(const v2f*)&tile[l16 * KV_STRIDE + kk * 4 + 2 * hf];
            const float* bp = bp0 + kk * 4 + 2 * hf;
            #pragma unroll
            for (int nt = 0; nt < 8; ++nt)
                acc[nt] = wmma4(a, *(const v2f*)(bp + (size_t)nt * 16 * L_), acc[nt]);
        }
        #pragma unroll
        for (int nt = 0; nt < 8; ++nt) {
            int n = wv * 128 + nt * 16 + l16;
            float bias = bk[n];
            #pragma unroll
            for (int i = 0; i < 8; ++i) {
                int row = m0 + i + 8 * hf;
                kout[((size_t)(b * M_ + row)) * E_ + n] = acc[nt][i] + bias;
            }
        }
    }
    // ---- v projection (stored transposed: vT[b][e][m]) ----
    #pragma unroll
    for (int i = 0; i < 8; ++i) acc[i] = vzero8();
    {
        const float* bp0 = WvT + (size_t)(wv * 128 + l16) * L_;
        for (int kk = 0; kk < 128; ++kk) {
            v2f a = *(const v2f*)&tile[l16 * KV_STRIDE + kk * 4 + 2 * hf];
            const float* bp = bp0 + kk * 4 + 2 * hf;
            #pragma unroll
            for (int nt = 0; nt < 8; ++nt)
                acc[nt] = wmma4(a, *(const v2f*)(bp + (size_t)nt * 16 * L_), acc[nt]);
        }
        #pragma unroll
        for (int nt = 0; nt < 8; ++nt) {
            int n = wv * 128 + nt * 16 + l16;   // e index
            float bias = bv[n];
            #pragma unroll
            for (int i = 0; i < 8; ++i) {
                int row = m0 + i + 8 * hf;      // m index
                vT[((size_t)(b * E_ + n)) * M_ + row] = acc[nt][i] + bias;
            }
        }
    }
}

// ---------------------------------------------------------------------------
// Fused: q = emb @ Wq + bq (kept in regs/LDS), w = (q @ k^T) * scale -> wbuf
// grid = B * (T/16) = 2048 blocks, 256 threads (8 waves)
// ---------------------------------------------------------------------------
#define QW_STRIDE 260   // 256 + 4 pad
__global__ __launch_bounds__(256) void k_qw(
        const float* __restrict__ emb,
        const float* __restrict__ WqT, const float* __restrict__ bq,
        const float* __restrict__ kbuf, float* __restrict__ wbuf) {
    __shared__ float tile[16 * QW_STRIDE];   // 16.6KB, reused emb->q
    const int b  = blockIdx.x >> 8;
    const int t0 = (blockIdx.x & 255) * 16;
    const int tid = threadIdx.x;
    const int lane = tid & 31;
    const int wv   = tid >> 5;
    const int l16  = lane & 15;
    const int hf   = lane >> 4;

    const float* embBase = emb + ((size_t)(b * T_ + t0)) * E_;

    // ---- phase 1: q slice (wave owns 128 of the 1024 output cols), K chunked
    v8f qacc[8];
    #pragma unroll
    for (int i = 0; i < 8; ++i) qacc[i] = vzero8();

    for (int kc = 0; kc < 4; ++kc) {
        __syncthreads();
        #pragma unroll
        for (int it = 0; it < 16; ++it) {
            int flat = tid + it * 256;
            int r = flat >> 8, c = flat & 255;
            tile[r * QW_STRIDE + c] = embBase[(size_t)r * E_ + kc * 256 + c];
        }
        __syncthreads();
        const float* bp0 = WqT + (size_t)(wv * 128 + l16) * E_ + kc * 256;
        for (int kk = 0; kk < 64; ++kk) {
            v2f a = *(const v2f*)&tile[l16 * QW_STRIDE + kk * 4 + 2 * hf];
            const float* bp = bp0 + kk * 4 + 2 * hf;
            #pragma unroll
            for (int nt = 0; nt < 8; ++nt)
                qacc[nt] = wmma4(a, *(const v2f*)(bp + (size_t)nt * 16 * E_), qacc[nt]);
        }
    }

    // ---- phase 2: w = q @ k^T (K = E = 1024, chunked via LDS re-stage of q)
    v8f wacc[2];
    wacc[0] = vzero8();
    wacc[1] = vzero8();
    const float* kBase = kbuf + (size_t)b * M_ * E_;

    for (int kc = 0; kc < 4; ++kc) {
        __syncthreads();
        if ((wv >> 1) == kc) {   // the two waves owning these 256 q columns
            int colbase = (wv & 1) * 128;
            #pragma unroll
            for (int nt = 0; nt < 8; ++nt) {
                int nglob = wv * 128 + nt * 16 + l16;
                float bias = bq[nglob];
                int lcol = colbase + nt * 16 + l16;
                #pragma unroll
                for (int i = 0; i < 8; ++i)
                    tile[(i + 8 * hf) * QW_STRIDE + lcol] = qacc[nt][i] + bias;
            }
        }
        __syncthreads();
        const float* bp0 = kBase + (size_t)(wv * 32 + l16) * E_ + kc * 256;
        for (int kk = 0; kk < 64; ++kk) {
            v2f a = *(const v2f*)&tile[l16 * QW_STRIDE + kk * 4 + 2 * hf];
            const float* bp = bp0 + kk * 4 + 2 * hf;
            wacc[0] = wmma4(a, *(const v2f*)(bp), wacc[0]);
            wacc[1] = wmma4(a, *(const v2f*)(bp + (size_t)16 * E_), wacc[1]);
        }
    }

    const float scale = 0.03125f;   // 1/sqrt(1024)
    float* wrow = wbuf + ((size_t)(b * T_ + t0)) * M_;
    #pragma unroll
    for (int nt = 0; nt < 2; ++nt) {
        int m = wv * 32 + nt * 16 + l16;
        #pragma unroll
        for (int i = 0; i < 8; ++i)
            wrow[(size_t)(i + 8 * hf) * M_ + m] = wacc[nt][i] * scale;
    }
}

// ---------------------------------------------------------------------------
// Column softmax stats over T (axis=1): online (max, sum-exp) partials
// ---------------------------------------------------------------------------
__global__ __launch_bounds__(256) void k_stats1(
        const float* __restrict__ wbuf, float* __restrict__ pmax,
        float* __restrict__ psum) {
    const int g = blockIdx.x;        // B*16
    const int b = g >> 4, sl = g & 15;
    const int m = threadIdx.x;
    const float* base = wbuf + ((size_t)(b * T_ + sl * 256)) * M_ + m;
    float mx = -3.4e38f, sm = 0.f;
    for (int t = 0; t < 256; ++t) {
        float v = base[(size_t)t * M_];
        float nmx = fmaxf(mx, v);
        sm = sm * __expf(mx - nmx) + __expf(v - nmx);
        mx = nmx;
    }
    pmax[g * 256 + m] = mx;
    psum[g * 256 + m] = sm;
}

__global__ __launch_bounds__(256) void k_stats2(
        const float* __restrict__ pmax, const float* __restrict__ psum,
        float* __restrict__ fmax, float* __restrict__ finv) {
    const int b = blockIdx.x;
    const int m = threadIdx.x;
    float mx = -3.4e38f;
    #pragma unroll
    for (int s = 0; s < 16; ++s) mx = fmaxf(mx, pmax[(b * 16 + s) * 256 + m]);
    float sm = 0.f;
    #pragma unroll
    for (int s = 0; s < 16; ++s)
        sm += psum[(b * 16 + s) * 256 + m] * __expf(pmax[(b * 16 + s) * 256 + m] - mx);
    fmax[b * 256 + m] = mx;
    finv[b * 256 + m] = 1.0f / sm;
}

// ---------------------------------------------------------------------------
// pred = softmax_cols(w) @ v ; x = pred + emb ; LayerNorm(E) -> out
// grid = B * (T/16) = 2048 blocks, 256 threads (8 waves)
// B-operand reads vT[b][e][m]: contiguous in m (= K dim) -> global_load_b64
// ---------------------------------------------------------------------------
__global__ __launch_bounds__(256) void k_out(
        const float* __restrict__ wbuf, const float* __restrict__ vT,
        const float* __restrict__ emb,
        const float* __restrict__ fmax, const float* __restrict__ finv,
        const float* __restrict__ gamma, const float* __restrict__ beta,
        float* __restrict__ out) {
    __shared__ float wn[16 * QW_STRIDE];   // normalized softmax weights tile
    __shared__ float wsumA[8][16], wsumB[8][16];
    __shared__ float rsum[16], rsumsq[16];
    const int b  = blockIdx.x >> 8;
    const int t0 = (blockIdx.x & 255) * 16;
    const int tid = threadIdx.x;
    const int lane = tid & 31;
    const int wv   = tid >> 5;
    const int l16  = lane & 15;
    const int hf   = lane >> 4;

    // stage softmax-normalized w tile (columns normalized by per-(b,m) stats)
    #pragma unroll
    for (int it = 0; it < 16; ++it) {
        int flat = tid + it * 256;
        int r = flat >> 8, c = flat & 255;
        float v = wbuf[((size_t)(b * T_ + t0 + r)) * M_ + c];
        wn[r * QW_STRIDE + c] = __expf(v - fmax[b * 256 + c]) * finv[b * 256 + c];
    }
    __syncthreads();

    // GEMM: [16 x 256] @ [256 x 1024], wave owns 128 output cols
    v8f acc[8];
    #pragma unroll
    for (int i = 0; i < 8; ++i) acc[i] = vzero8();
    const float* vb = vT + (size_t)b * E_ * M_;
    {
        const float* bp0 = vb + (size_t)(wv * 128 + l16) * M_;
        for (int kk = 0; kk < 64; ++kk) {
            v2f a = *(const v2f*)&wn[l16 * QW_STRIDE + kk * 4 + 2 * hf];
            const float* bp = bp0 + kk * 4 + 2 * hf;
            #pragma unroll
            for (int nt = 0; nt < 8; ++nt)
                acc[nt] = wmma4(a, *(const v2f*)(bp + (size_t)nt * 16 * M_), acc[nt]);
        }
    }

    // residual add + per-lane LayerNorm partials
    float ps[8], pq[8];
    #pragma unroll
    for (int i = 0; i < 8; ++i) { ps[i] = 0.f; pq[i] = 0.f; }
    const float* embB = emb + ((size_t)(b * T_ + t0)) * E_;
    #pragma unroll
    for (int nt = 0; nt < 8; ++nt) {
        int n = wv * 128 + nt * 16 + l16;
        #pragma unroll
        for (int i = 0; i < 8; ++i) {
            float x = acc[nt][i] + embB[(size_t)(i + 8 * hf) * E_ + n];
            acc[nt][i] = x;
            ps[i] += x;
            pq[i] += x * x;
        }
    }
    // deterministic reduction: shuffle within 16-lane half, then fixed-order
    #pragma unroll
    for (int i = 0; i < 8; ++i) {
        float s = ps[i], q = pq[i];
        #pragma unroll
        for (int off = 8; off >= 1; off >>= 1) {
            s += __shfl_xor(s, off, 32);
            q += __shfl_xor(q, off, 32);
        }
        if (l16 == 0) {
            wsumA[wv][hf * 8 + i] = s;
            wsumB[wv][hf * 8 + i] = q;
        }
    }
    __syncthreads();
    if (tid < 16) {
        float s = 0.f, q = 0.f;
        #pragma unroll
        for (int w2 = 0; w2 < 8; ++w2) { s += wsumA[w2][tid]; q += wsumB[w2][tid]; }
        rsum[tid] = s;
        rsumsq[tid] = q;
    }
    __syncthreads();

    // normalize + affine + store
    #pragma unroll
    for (int nt = 0; nt < 8; ++nt) {
        int n = wv * 128 + nt * 16 + l16;
        float g = gamma[n], be = beta[n];
        #pragma unroll
        for (int i = 0; i < 8; ++i) {
            int r = i + 8 * hf;
            float mu  = rsum[r] * (1.0f / (float)E_);
            float var = rsumsq[r] * (1.0f / (float)E_) - mu * mu;
            float rs  = rsqrtf(var + EPS_);
            out[((size_t)(b * T_ + t0 + r)) * E_ + n] = (acc[nt][i] - mu) * rs * g + be;
        }
    }
}

// ---------------------------------------------------------------------------
extern "C" void kernel_launch(void* const* d_in, const int* in_sizes, int n_in,
                              void* d_out, int out_size, void* d_ws, size_t ws_size,
                              hipStream_t stream) {
    const float* emb    = (const float*)d_in[0];
    const float* latent = (const float*)d_in[1];
    const float* Wq     = (const float*)d_in[2];
    const float* bq     = (const float*)d_in[3];
    const float* Wk     = (const float*)d_in[4];
    const float* bk     = (const float*)d_in[5];
    const float* Wv     = (const float*)d_in[6];
    const float* bv     = (const float*)d_in[7];
    const float* gamma  = (const float*)d_in[8];
    const float* beta   = (const float*)d_in[9];
    float* out = (float*)d_out;

    // workspace layout (floats), ~59 MB total
    float* ws   = (float*)d_ws;
    float* WqT  = ws;                                     // E*E
    float* WkT  = WqT + (size_t)E_ * E_;                  // E*L
    float* WvT  = WkT + (size_t)E_ * L_;                  // E*L
    float* kbuf = WvT + (size_t)E_ * L_;                  // B*M*E
    float* vTb  = kbuf + (size_t)B_ * M_ * E_;            // B*E*M (transposed v)
    float* wbuf = vTb + (size_t)B_ * M_ * E_;             // B*T*M
    float* pmax = wbuf + (size_t)B_ * T_ * M_;            // B*16*256
    float* psum = pmax + (size_t)B_ * 16 * 256;           // B*16*256
    float* fmx  = psum + (size_t)B_ * 16 * 256;           // B*256
    float* finv = fmx  + (size_t)B_ * 256;                // B*256

    dim3 tb(32, 8);
    k_transpose<<<dim3(E_ / 32, E_ / 32), tb, 0, stream>>>(Wq, WqT, E_, E_);
    k_transpose<<<dim3(E_ / 32, L_ / 32), tb, 0, stream>>>(Wk, WkT, L_, E_);
    k_transpose<<<dim3(E_ / 32, L_ / 32), tb, 0, stream>>>(Wv, WvT, L_, E_);

    k_kv<<<B_ * (M_ / 16), 256, 0, stream>>>(latent, WkT, bk, WvT, bv, kbuf, vTb);
    k_qw<<<B_ * (T_ / 16), 256, 0, stream>>>(emb, WqT, bq, kbuf, wbuf);
    k_stats1<<<B_ * 16, 256, 0, stream>>>(wbuf, pmax, psum);
    k_stats2<<<B_, 256, 0, stream>>>(pmax, psum, fmx, finv);
    k_out<<<B_ * (T_ / 16), 256, 0, stream>>>(wbuf, vTb, emb, fmx, finv,
                                              gamma, beta, out);
}